// NeuralDivingNet_91250875171069
// MI455X (gfx1250) — compile-verified
//
#include <hip/hip_runtime.h>
#include <stdint.h>

#define HD    128
#define FEATN 16
#define NVARS 50000
#define NINT  20000
#define KCAT  8

typedef float v2f __attribute__((ext_vector_type(2)));
typedef float v8f __attribute__((ext_vector_type(8)));
typedef unsigned int u32x4 __attribute__((ext_vector_type(4)));
typedef int i32x8 __attribute__((ext_vector_type(8)));
typedef int i32x4 __attribute__((ext_vector_type(4)));

#if __has_builtin(__builtin_amdgcn_tensor_load_to_lds) && \
    __has_builtin(__builtin_amdgcn_s_wait_tensorcnt)
#define CDNA5_HAS_TDM 1
#else
#define CDNA5_HAS_TDM 0
#endif

__device__ __forceinline__ v8f wmma_f32_k4(v2f a, v2f b, v8f c) {
  // D = A(16x4,f32) * B(4x16,f32) + C(16x16,f32)
  return __builtin_amdgcn_wmma_f32_16x16x4_f32(false, a, false, b, (short)0, c,
                                               false, false);
}

#if CDNA5_HAS_TDM
// Issue a 1-D TDM copy of nflt fp32 elements (nflt <= 65535) global -> LDS.
// Per-wave instruction: call from ONE wave only, EXEC uniform.
__device__ __forceinline__ void tdm_load_f32(void* ldsdst, const void* gsrc,
                                             unsigned nflt) {
  unsigned laddr = (unsigned)(uintptr_t)ldsdst;  // LDS aperture: low 32 bits
  unsigned long long ga = (unsigned long long)(uintptr_t)gsrc;
  u32x4 g0;
  g0[0] = 1u;                                   // count=1 (valid descriptor)
  g0[1] = laddr;                                // lds_addr (bytes)
  g0[2] = (unsigned)(ga & 0xffffffffu);         // global_addr[31:0]
  g0[3] = (unsigned)((ga >> 32) & 0x01ffffffu)  // global_addr[56:32]
          | 0x80000000u;                        // type=2 ("image")
  i32x8 g1;
  g1[0] = (int)(2u << 16);                      // data_size=4B; mask=0
  g1[1] = (int)((nflt & 0xffffu) << 16);        // tensor_dim0[15:0]
  g1[2] = (int)((nflt >> 16) | (1u << 16));     // tensor_dim0[31:16], dim1=1
  g1[3] = (int)((nflt & 0xffffu) << 16);        // tile_dim0 = nflt
  g1[4] = 1;                                    // tile_dim1 = 1
  g1[5] = (int)nflt;                            // tensor_dim0_stride
  g1[6] = 0;
  g1[7] = 0;
  i32x4 z = {};
#if defined(__clang_major__) && (__clang_major__ >= 23)
  i32x8 z8 = {};
  __builtin_amdgcn_tensor_load_to_lds(g0, g1, z, z, z8, 0);
#else
  __builtin_amdgcn_tensor_load_to_lds(g0, g1, z, z, 0);
#endif
  __builtin_amdgcn_s_wait_tensorcnt(0);
}
#endif

// ---------------------------------------------------------------------------
// Weight repack: Wp[k2*N+c] = (W[2k2][c], W[2k2+1][c])  -> B frag = one b64
// ---------------------------------------------------------------------------
__global__ void repack_kernel(const float* __restrict__ W,
                              float* __restrict__ Wp, int K, int N) {
  int i = blockIdx.x * blockDim.x + threadIdx.x;
  int total = (K >> 1) * N;
  if (i >= total) return;
  int k2 = i / N, c = i - k2 * N;
  Wp[2 * i + 0] = W[(size_t)(2 * k2 + 0) * N + c];
  Wp[2 * i + 1] = W[(size_t)(2 * k2 + 1) * N + c];
}

// ---------------------------------------------------------------------------
// Degree / normalization kernels
// ---------------------------------------------------------------------------
__global__ void deg_init_kernel(float* __restrict__ deg, int n) {
  int i = blockIdx.x * blockDim.x + threadIdx.x;
  if (i < n) deg[i] = 1.0f;
}

__global__ void deg_edge_kernel(const long long* __restrict__ dst,
                                float* __restrict__ deg, long long nedges) {
  long long e = (long long)blockIdx.x * blockDim.x + threadIdx.x;
  if (e < nedges) atomicAdd(&deg[(int)dst[e]], 1.0f);
}

__global__ void deg_finish_kernel(float* __restrict__ deg_then_dinv,
                                  float* __restrict__ self_norm, int n) {
  int i = blockIdx.x * blockDim.x + threadIdx.x;
  if (i < n) {
    float d = deg_then_dinv[i];
    self_norm[i] = 1.0f / d;
    deg_then_dinv[i] = rsqrtf(d);
  }
}

// ---------------------------------------------------------------------------
// Encoder: y = relu(x @ W1 + b1) @ W2 + b2    (x: nrows x 16)
// One wave per 16-row tile; activations staged in LDS. B from packed weights.
// ---------------------------------------------------------------------------
__global__ void enc_kernel(const float* __restrict__ xin,  // [nrows x 16]
                           const v2f* __restrict__ W1p,    // packed [16 x 128]
                           const float* __restrict__ b1,
                           const v2f* __restrict__ W2p,    // packed [128 x 128]
                           const float* __restrict__ b2,
                           float* __restrict__ hout,       // [nrows x 128]
                           int nrows) {
  __shared__ float lds[8][16 * HD];
  const int wave = threadIdx.x >> 5;
  const int lane = threadIdx.x & 31;
  const int half = lane >> 4;
  const int ln   = lane & 15;
  const int nblk = (nrows + 15) >> 4;
  int blk = blockIdx.x * 8 + wave;
  const bool valid = blk < nblk;
  if (!valid) blk = nblk - 1;
  const int row0 = blk * 16;
  float* myl = lds[wave];

  // Stage 1: x(16x16) @ W1(16x128), +b1, relu -> LDS
#pragma unroll
  for (int t = 0; t < 8; ++t) {
    const int cn = t * 16;
    v8f acc = {};
#pragma unroll
    for (int kk = 0; kk < 4; ++kk) {
      const int kb = kk * 4 + half * 2;
      v2f a = *(const v2f*)(xin + (size_t)(row0 + ln) * FEATN + kb);
      v2f b = W1p[(size_t)(2 * kk + half) * HD + cn + ln];
      acc = wmma_f32_k4(a, b, acc);
    }
    const float bias = b1[cn + ln];
#pragma unroll
    for (int r = 0; r < 8; ++r) {
      float v = acc[r] + bias;
      myl[(r + half * 8) * HD + cn + ln] = v > 0.0f ? v : 0.0f;
    }
  }
  __syncthreads();

  // Stage 2: LDS(16x128) @ W2(128x128) + b2 -> hout
#pragma unroll
  for (int t = 0; t < 8; ++t) {
    const int cn = t * 16;
    v8f acc = {};
#pragma unroll
    for (int kk = 0; kk < 32; ++kk) {
      const int kb = kk * 4 + half * 2;
      v2f a = *(const v2f*)(myl + ln * HD + kb);
      v2f b = W2p[(size_t)(2 * kk + half) * HD + cn + ln];
      acc = wmma_f32_k4(a, b, acc);
    }
    if (valid) {
      const float bias = b2[cn + ln];
#pragma unroll
      for (int r = 0; r < 8; ++r) {
        int row = row0 + r + half * 8;
        if (row < nrows) hout[(size_t)row * HD + cn + ln] = acc[r] + bias;
      }
    }
  }
}

// ---------------------------------------------------------------------------
// hw = h @ conv_W ; agg = hw * self_norm + conv_b
// Packed conv weights staged to LDS via Tensor Data Mover; B frags = ds b64.
// ---------------------------------------------------------------------------
__global__ void hw_agg_init_kernel(const float* __restrict__ h,   // [N x 128]
                                   const v2f* __restrict__ Wp,    // packed W
                                   const float* __restrict__ bias,
                                   const float* __restrict__ self_norm,
                                   float* __restrict__ hw,
                                   float* __restrict__ agg,
                                   int nrows) {
  __shared__ v2f wlds[64 * HD];  // 64KB: packed 128x128 weight tile
  const int wave = threadIdx.x >> 5;
  const int lane = threadIdx.x & 31;
  const int half = lane >> 4;
  const int ln   = lane & 15;
  const int nblk = (nrows + 15) >> 4;
  int blk = blockIdx.x * 8 + wave;
  const bool valid = blk < nblk;
  if (!valid) blk = nblk - 1;
  const int row0 = blk * 16;

#if CDNA5_HAS_TDM
  if (threadIdx.x < 32) tdm_load_f32(wlds, Wp, HD * HD);
#else
  for (int i = threadIdx.x; i < 64 * HD; i += 256) wlds[i] = Wp[i];
#endif
  __builtin_prefetch(h + (size_t)(row0 + ln) * HD, 0, 0);
  __syncthreads();

#pragma unroll
  for (int t = 0; t < 8; ++t) {
    const int cn = t * 16;
    v8f acc = {};
#pragma unroll
    for (int kk = 0; kk < 32; ++kk) {
      const int kb = kk * 4 + half * 2;
      v2f a = *(const v2f*)(h + (size_t)(row0 + ln) * HD + kb);
      v2f b = wlds[(2 * kk + half) * HD + cn + ln];
      acc = wmma_f32_k4(a, b, acc);
    }
    if (valid) {
      const float bb = bias[cn + ln];
#pragma unroll
      for (int r = 0; r < 8; ++r) {
        int row = row0 + r + half * 8;
        if (row < nrows) {
          float v = acc[r];
          hw[(size_t)row * HD + cn + ln] = v;
          agg[(size_t)row * HD + cn + ln] = v * self_norm[row] + bb;
        }
      }
    }
  }
}

// ---------------------------------------------------------------------------
// Edge scatter: agg[dst] += hw[src] * (dinv[src]*dinv[dst])
// One thread per (edge, 4 cols) -> global_atomic_add_f32, L2-resident.
// ---------------------------------------------------------------------------
__global__ void edge_scatter_kernel(const long long* __restrict__ src,
                                    const long long* __restrict__ dst,
                                    const float* __restrict__ dinv,
                                    const float* __restrict__ hw,
                                    float* __restrict__ agg,
                                    long long nedges) {
  long long tid = (long long)blockIdx.x * blockDim.x + threadIdx.x;
  long long e = tid >> 5;
  if (e >= nedges) return;
  const int cg = (int)(tid & 31);
  const int s = (int)src[e];
  const int d = (int)dst[e];
  const float en = dinv[s] * dinv[d];
  const float4 v = *(const float4*)(hw + (size_t)s * HD + cg * 4);
  float* ap = agg + (size_t)d * HD + cg * 4;
  atomicAdd(ap + 0, v.x * en);
  atomicAdd(ap + 1, v.y * en);
  atomicAdd(ap + 2, v.z * en);
  atomicAdd(ap + 3, v.w * en);
}

// ---------------------------------------------------------------------------
// Refine: h += relu( relu(agg) @ W1 + b1 ) @ W2 + b2
// ---------------------------------------------------------------------------
__global__ void refine_kernel(const float* __restrict__ agg,  // [N x 128]
                              const v2f* __restrict__ W1p,
                              const float* __restrict__ b1,
                              const v2f* __restrict__ W2p,
                              const float* __restrict__ b2,
                              float* __restrict__ h,          // in/out
                              int nrows) {
  __shared__ float lds[8][16 * HD];
  const int wave = threadIdx.x >> 5;
  const int lane = threadIdx.x & 31;
  const int half = lane >> 4;
  const int ln   = lane & 15;
  const int nblk = (nrows + 15) >> 4;
  int blk = blockIdx.x * 8 + wave;
  const bool valid = blk < nblk;
  if (!valid) blk = nblk - 1;
  const int row0 = blk * 16;
  float* myl = lds[wave];

  // Stage 1: relu(agg)(16x128) @ W1 + b1, relu -> LDS
#pragma unroll
  for (int t = 0; t < 8; ++t) {
    const int cn = t * 16;
    v8f acc = {};
#pragma unroll
    for (int kk = 0; kk < 32; ++kk) {
      const int kb = kk * 4 + half * 2;
      v2f a = *(const v2f*)(agg + (size_t)(row0 + ln) * HD + kb);
      a.x = a.x > 0.0f ? a.x : 0.0f;
      a.y = a.y > 0.0f ? a.y : 0.0f;
      v2f b = W1p[(size_t)(2 * kk + half) * HD + cn + ln];
      acc = wmma_f32_k4(a, b, acc);
    }
    const float bias = b1[cn + ln];
#pragma unroll
    for (int r = 0; r < 8; ++r) {
      float v = acc[r] + bias;
      myl[(r + half * 8) * HD + cn + ln] = v > 0.0f ? v : 0.0f;
    }
  }
  __syncthreads();

  // Stage 2: LDS @ W2 + b2 + h(residual) -> h
#pragma unroll
  for (int t = 0; t < 8; ++t) {
    const int cn = t * 16;
    v8f acc = {};
#pragma unroll
    for (int kk = 0; kk < 32; ++kk) {
      const int kb = kk * 4 + half * 2;
      v2f a = *(const v2f*)(myl + ln * HD + kb);
      v2f b = W2p[(size_t)(2 * kk + half) * HD + cn + ln];
      acc = wmma_f32_k4(a, b, acc);
    }
    if (valid) {
      const float bias = b2[cn + ln];
#pragma unroll
      for (int r = 0; r < 8; ++r) {
        int row = row0 + r + half * 8;
        if (row < nrows) {
          size_t idx = (size_t)row * HD + cn + ln;
          h[idx] = h[idx] + acc[r] + bias;
        }
      }
    }
  }
}

// ---------------------------------------------------------------------------
// Bernoulli head: out = relu(vh @ bh_W1 + b1) @ bh_W2 + b2   (128->64->1)
// Packed bh_W1 staged to LDS via TDM; stage2 = VALU dot.
// ---------------------------------------------------------------------------
__global__ void bern_kernel(const float* __restrict__ h,   // [nrows x 128]
                            const v2f* __restrict__ W1p,   // packed [128 x 64]
                            const float* __restrict__ b1,
                            const float* __restrict__ W2,  // [64 x 1]
                            const float* __restrict__ b2,
                            float* __restrict__ out, int nrows) {
  __shared__ float hid[8][16 * 64];  // 32KB
  __shared__ v2f wlds[64 * 64];      // 32KB packed weights
  const int wave = threadIdx.x >> 5;
  const int lane = threadIdx.x & 31;
  const int half = lane >> 4;
  const int ln   = lane & 15;
  const int nblk = (nrows + 15) >> 4;
  int blk = blockIdx.x * 8 + wave;
  const bool valid = blk < nblk;
  if (!valid) blk = nblk - 1;
  const int row0 = blk * 16;
  float* myl = hid[wave];

#if CDNA5_HAS_TDM
  if (threadIdx.x < 32) tdm_load_f32(wlds, W1p, HD * 64);
#else
  for (int i = threadIdx.x; i < 64 * 64; i += 256) wlds[i] = W1p[i];
#endif
  __syncthreads();

#pragma unroll
  for (int t = 0; t < 4; ++t) {
    const int cn = t * 16;
    v8f acc = {};
#pragma unroll
    for (int kk = 0; kk < 32; ++kk) {
      const int kb = kk * 4 + half * 2;
      v2f a = *(const v2f*)(h + (size_t)(row0 + ln) * HD + kb);
      v2f b = wlds[(2 * kk + half) * 64 + cn + ln];
      acc = wmma_f32_k4(a, b, acc);
    }
    const float bias = b1[cn + ln];
#pragma unroll
    for (int r = 0; r < 8; ++r) {
      float v = acc[r] + bias;
      myl[(r + half * 8) * 64 + cn + ln] = v > 0.0f ? v : 0.0f;
    }
  }
  __syncthreads();

  if (half == 0) {  // lanes 0..15, one row each
    float s = 0.0f;
#pragma unroll
    for (int k = 0; k < 64; ++k) s += myl[ln * 64 + k] * W2[k];
    int row = row0 + ln;
    if (valid && row < nrows) out[row] = s + b2[0];
  }
}

// ---------------------------------------------------------------------------
// Categorical head: cat[n,k] = sum_h vh[n,h] * cat_W[n,h,k] + cat_b[n,k]
// ---------------------------------------------------------------------------
__global__ void cat_kernel(const float* __restrict__ h,
                           const float* __restrict__ catW,  // [NINT x 128 x 8]
                           const float* __restrict__ catB,  // [NINT x 8]
                           float* __restrict__ out, int nint) {
  int tid = blockIdx.x * blockDim.x + threadIdx.x;
  int n = tid >> 3;
  int k = tid & 7;
  if (n >= nint) return;
  float s = catB[n * KCAT + k];
  const float* hv = h + (size_t)n * HD;
  const float* wv = catW + (size_t)n * HD * KCAT + k;
#pragma unroll 8
  for (int hh = 0; hh < HD; ++hh) s += hv[hh] * wv[(size_t)hh * KCAT];
  out[n * KCAT + k] = s;
}

// ---------------------------------------------------------------------------
extern "C" void kernel_launch(void* const* d_in, const int* in_sizes, int n_in,
                              void* d_out, int out_size, void* d_ws, size_t ws_size,
                              hipStream_t stream) {
  const float*     x  = (const float*)d_in[0];
  const long long* ei = (const long long*)d_in[1];  // int64 edge_index [2 x E]
  const int N = in_sizes[0] / FEATN;                // 100000
  const long long E = in_sizes[1] / 2;              // 640000
  const int NV = NVARS;

  const float* ve_W1 = (const float*)d_in[3];
  const float* ve_b1 = (const float*)d_in[4];
  const float* ve_W2 = (const float*)d_in[5];
  const float* ve_b2 = (const float*)d_in[6];
  const float* ce_W1 = (const float*)d_in[7];
  const float* ce_b1 = (const float*)d_in[8];
  const float* ce_W2 = (const float*)d_in[9];
  const float* ce_b2 = (const float*)d_in[10];
  const float* conv_W = (const float*)d_in[11];  // [4 x 128 x 128]
  const float* conv_b = (const float*)d_in[12];
  const float* ref_W1 = (const float*)d_in[13];
  const float* ref_b1 = (const float*)d_in[14];
  const float* ref_W2 = (const float*)d_in[15];
  const float* ref_b2 = (const float*)d_in[16];
  const float* bh_W1  = (const float*)d_in[17];
  const float* bh_b1  = (const float*)d_in[18];
  const float* bh_W2  = (const float*)d_in[19];
  const float* bh_b2  = (const float*)d_in[20];
  const float* cat_W  = (const float*)d_in[21];
  const float* cat_b  = (const float*)d_in[22];

  float* ws        = (float*)d_ws;
  float* dinv      = ws;                      // [N]
  float* self_norm = ws + N;                  // [N]
  float* h         = ws + 2 * (size_t)N;      // [N x 128]
  float* hw        = h  + (size_t)N * HD;     // [N x 128]
  float* agg       = hw + (size_t)N * HD;     // [N x 128]
  float* wpk       = agg + (size_t)N * HD;    // packed weights
  float* ve_W1p = wpk;                  // 2048
  float* ce_W1p = ve_W1p + 2048;        // 2048
  float* ve_W2p = ce_W1p + 2048;        // 16384
  float* ce_W2p = ve_W2p + 16384;       // 16384
  float* conv_Wp = ce_W2p + 16384;      // 4*16384
  float* ref_W1p = conv_Wp + 4 * 16384; // 4*16384
  float* ref_W2p = ref_W1p + 4 * 16384; // 4*16384
  float* bh_W1p  = ref_W2p + 4 * 16384; // 8192

  const long long* src = ei;
  const long long* dst = ei + E;

  // ---- weight repacking (B fragments as b64) ----
  repack_kernel<<<(1024 + 255) / 256, 256, 0, stream>>>(ve_W1, ve_W1p, 16, HD);
  repack_kernel<<<(1024 + 255) / 256, 256, 0, stream>>>(ce_W1, ce_W1p, 16, HD);
  repack_kernel<<<(8192 + 255) / 256, 256, 0, stream>>>(ve_W2, ve_W2p, HD, HD);
  repack_kernel<<<(8192 + 255) / 256, 256, 0, stream>>>(ce_W2, ce_W2p, HD, HD);
  for (int l = 0; l < 4; ++l) {
    repack_kernel<<<32, 256, 0, stream>>>(conv_W + (size_t)l * HD * HD,
                                          conv_Wp + (size_t)l * HD * HD, HD, HD);
    repack_kernel<<<32, 256, 0, stream>>>(ref_W1 + (size_t)l * HD * HD,
                                          ref_W1p + (size_t)l * HD * HD, HD, HD);
    repack_kernel<<<32, 256, 0, stream>>>(ref_W2 + (size_t)l * HD * HD,
                                          ref_W2p + (size_t)l * HD * HD, HD, HD);
  }
  repack_kernel<<<(4096 + 255) / 256, 256, 0, stream>>>(bh_W1, bh_W1p, HD, 64);

  // ---- degrees & normalization ----
  deg_init_kernel<<<(N + 255) / 256, 256, 0, stream>>>(dinv, N);
  deg_edge_kernel<<<(int)((E + 255) / 256), 256, 0, stream>>>(dst, dinv, E);
  deg_finish_kernel<<<(N + 255) / 256, 256, 0, stream>>>(dinv, self_norm, N);

  // ---- encoders ----
  const int nblkV = (NV + 15) / 16;
  const int nblkC = ((N - NV) + 15) / 16;
  enc_kernel<<<(nblkV + 7) / 8, 256, 0, stream>>>(
      x, (const v2f*)ve_W1p, ve_b1, (const v2f*)ve_W2p, ve_b2, h, NV);
  enc_kernel<<<(nblkC + 7) / 8, 256, 0, stream>>>(
      x + (size_t)NV * FEATN, (const v2f*)ce_W1p, ce_b1, (const v2f*)ce_W2p,
      ce_b2, h + (size_t)NV * HD, N - NV);

  // ---- GCN layers ----
  const int nblkN = (N + 15) / 16;
  for (int l = 0; l < 4; ++l) {
    hw_agg_init_kernel<<<(nblkN + 7) / 8, 256, 0, stream>>>(
        h, (const v2f*)(conv_Wp + (size_t)l * HD * HD), conv_b + l * HD,
        self_norm, hw, agg, N);
    long long tot = E * 32;
    edge_scatter_kernel<<<(int)((tot + 255) / 256), 256, 0, stream>>>(
        src, dst, dinv, hw, agg, E);
    refine_kernel<<<(nblkN + 7) / 8, 256, 0, stream>>>(
        agg, (const v2f*)(ref_W1p + (size_t)l * HD * HD), ref_b1 + l * HD,
        (const v2f*)(ref_W2p + (size_t)l * HD * HD), ref_b2 + l * HD, h, N);
  }

  // ---- heads ----
  float* out = (float*)d_out;
  bern_kernel<<<(nblkV + 7) / 8, 256, 0, stream>>>(
      h, (const v2f*)bh_W1p, bh_b1, bh_W2, bh_b2, out, NV);
  cat_kernel<<<(NINT * KCAT + 255) / 256, 256, 0, stream>>>(
      h, cat_W, cat_b, out + NV, NINT);
}